// PostProcess_62775241998895
// MI455X (gfx1250) — compile-verified
//
#include <hip/hip_runtime.h>

#define L 512
#define LL (L * L)

typedef __attribute__((ext_vector_type(2))) float v2f;
typedef __attribute__((ext_vector_type(8))) float v8f;

// ---------------------------------------------------------------------------
// Kernel 0: zero the output (greedy only writes accepted pairs).
// ---------------------------------------------------------------------------
__global__ void zero_out_kernel(float* __restrict__ out) {
    int i = blockIdx.x * blockDim.x + threadIdx.x;
    if (i < LL) out[i] = 0.0f;
}

// ---------------------------------------------------------------------------
// Kernel 1: per-position base value from argmax over 4 channels (column 0).
// feat layout (1,8,L,2): feat[ch,pos,0] at ((ch*L)+pos)*2.
// Strict '>' keeps the first max, matching jnp.argmax.
// ---------------------------------------------------------------------------
__global__ void pairs_kernel(const float* __restrict__ feat,
                             float* __restrict__ pairs) {
    int pos = blockIdx.x * blockDim.x + threadIdx.x;
    if (pos >= L) return;
    float bv = feat[(0 * L + pos) * 2];
    int bi = 0;
#pragma unroll
    for (int ch = 1; ch < 4; ++ch) {
        float v = feat[(ch * L + pos) * 2];
        if (v > bv) { bv = v; bi = ch; }
    }
    const float BASES[4] = {2.0f, 3.0f, 5.0f, 7.0f};
    pairs[pos] = BASES[bi];
}

// ---------------------------------------------------------------------------
// Kernel 2: masked score matrix via WMMA rank-1 outer product.
// One wave per 16x16 tile: V_WMMA_F32_16X16X4_F32 with only K=0 populated
// computes pairs[m]*pairs[n]; membership test {14,15,35} + triu(k=5) gate.
// Unconditional con loads + selects (no per-element exec diamonds).
// Tiles fully below the band short-circuit to zero stores (uniform branch).
// ---------------------------------------------------------------------------
__global__ void mask_score_kernel(const float* __restrict__ con,
                                  const float* __restrict__ pairs,
                                  float* __restrict__ c) {
    const int tileN = blockIdx.x * 16;  // column tile
    const int tileM = blockIdx.y * 16;  // row tile
    const int lane  = threadIdx.x;      // 0..31 (wave32)

    const int n  = tileN + (lane & 15);
    const int mb = tileM + ((lane >> 4) << 3);

    // Entire tile below triu(k=5) band? (max col) - (min row) < 5 -> all zero.
    if (tileN + 15 - tileM < 5) {
#pragma unroll
        for (int v = 0; v < 8; ++v) c[(mb + v) * L + n] = 0.0f;
        return;
    }

    v2f a; a.x = 0.0f; a.y = 0.0f;
    v2f b; b.x = 0.0f; b.y = 0.0f;
    if (lane < 16) {
        a.x = pairs[tileM + lane];  // A: M=lane, K=0
        b.x = pairs[tileN + lane];  // B: K=0, N=lane
    }
    v8f acc = {};
    // D[m][n] = pairs[tileM+m] * pairs[tileN+n]  (EXEC all-1s here)
    v8f d = __builtin_amdgcn_wmma_f32_16x16x4_f32(
        /*neg_a=*/false, a, /*neg_b=*/false, b,
        /*c_mod=*/(short)0, acc, /*reuse_a=*/false, /*reuse_b=*/false);

    float cv[8];
#pragma unroll
    for (int v = 0; v < 8; ++v) cv[v] = con[(mb + v) * L + n];  // always in-bounds

#pragma unroll
    for (int v = 0; v < 8; ++v) {
        int   m     = mb + v;
        float prod  = d[v];
        bool  valid = (prod == 14.0f) | (prod == 15.0f) | (prod == 35.0f);
        c[m * L + n] = (valid && (n - m >= 5)) ? cv[v] : 0.0f;
    }
}

// ---------------------------------------------------------------------------
// Kernel 3: greedy matching with a lazy per-row max cache.
// rowBest[r] = argmax over FREE columns of c[r,:]. Removing a non-argmax
// column never changes a row's max, so a row is recomputed only when its
// cached argmax column gets matched. Rows whose max over free columns is
// <= 0 can never recover (c is static, columns only get removed) -> mark
// exhausted (-2) permanently. One thread per row; 512 threads.
// Tie-breaks match stable argsort(-flat): within a row, ascending col scan
// with strict '>' keeps the smallest col; across rows, (val desc, row asc)
// equals flat-index order since r*L dominates.
// ---------------------------------------------------------------------------
__global__ void greedy_kernel(const float* __restrict__ c,
                              float* __restrict__ out) {
    const int tid = threadIdx.x;            // 0..511
    __shared__ float rowVal[L];
    __shared__ int   rowCol[L];             // -1 = stale, -2 = exhausted
    __shared__ int   used[L];
    __shared__ float rv[L];
    __shared__ int   ri[L];
    __shared__ int   done;

    used[tid]   = 0;
    rowVal[tid] = 0.0f;
    rowCol[tid] = -1;
    if (tid == 0) done = 0;
    __syncthreads();

    for (int iter = 0; iter < L / 2; ++iter) {
        // --- Step 1: lazily recompute rows whose cached argmax col is gone.
        {
            const int r  = tid;
            const int rc = rowCol[r];
            if (!used[r] && rc != -2 && (rc == -1 || used[rc])) {
                float bv = 0.0f;
                int   bc = -2;
                const float* rowp = c + r * L;
                __builtin_prefetch(rowp + (r + 5), 0, 1);
                for (int col = r + 5; col < L; ++col) {   // band: c==0 elsewhere
                    if (used[col]) continue;
                    float v = rowp[col];
                    if (v > bv) { bv = v; bc = col; }
                }
                rowVal[r] = bv;
                rowCol[r] = bc;                            // -2 if nothing > 0
            }
        }
        __syncthreads();

        // --- Step 2: reduce over rows: max value, ties -> smaller row index.
        rv[tid] = (!used[tid] && rowCol[tid] >= 0) ? rowVal[tid] : 0.0f;
        ri[tid] = tid;
        __syncthreads();
        for (int s = L / 2; s > 0; s >>= 1) {
            if (tid < s) {
                float ov = rv[tid + s];
                int   oi = ri[tid + s];
                if (ov > rv[tid] || (ov == rv[tid] && oi < ri[tid])) {
                    rv[tid] = ov;
                    ri[tid] = oi;
                }
            }
            __syncthreads();
        }

        // --- Step 3: accept best pair or terminate.
        if (tid == 0) {
            if (rv[0] <= 0.0f) {
                done = 1;
            } else {
                int r   = ri[0];
                int col = rowCol[r];
                used[r]   = 1;
                used[col] = 1;
                out[r * L + col] = rv[0];
                out[col * L + r] = rv[0];
            }
        }
        __syncthreads();
        if (done) break;
    }
}

// ---------------------------------------------------------------------------
// Launch: zero out -> pairs -> WMMA masked scores -> lazy greedy matching.
// Workspace: pairs (L f32, padded to 4 KB) + score matrix c (1 MB).
// ---------------------------------------------------------------------------
extern "C" void kernel_launch(void* const* d_in, const int* in_sizes, int n_in,
                              void* d_out, int out_size, void* d_ws, size_t ws_size,
                              hipStream_t stream) {
    const float* con  = (const float*)d_in[0];  // (1, L, L) float32
    const float* feat = (const float*)d_in[1];  // (1, 8, L, 2) float32
    float* out   = (float*)d_out;               // (1, L, L) float32
    float* pairs = (float*)d_ws;                // L floats
    float* c     = pairs + 1024;                // L*L floats, 4 KB-aligned

    zero_out_kernel<<<(LL + 255) / 256, 256, 0, stream>>>(out);
    pairs_kernel<<<(L + 127) / 128, 128, 0, stream>>>(feat, pairs);

    dim3 tiles(L / 16, L / 16);                 // 32x32 tiles, 1 wave each
    mask_score_kernel<<<tiles, 32, 0, stream>>>(con, pairs, c);

    greedy_kernel<<<1, L, 0, stream>>>(c, out);
}